// IDloss_46995532153217
// MI455X (gfx1250) — compile-verified
//
#include <hip/hip_runtime.h>

// MI455X / gfx1250, wave32. Problem constants from the reference.
#define M_ANCH 16384
#define P_POS  3
#define N_NEG  63
#define D_DIM  128
#define EPS_V  1e-8f
#define NTILES (M_ANCH / 16)   // 1024 anchor tiles, one wave each

typedef __attribute__((ext_vector_type(2))) float v2f;
typedef __attribute__((ext_vector_type(8))) float v8f;

// One wave per 16-anchor tile.
//  B (4x16 f32 per K-chunk)  = 16 anchor columns, loaded once, reused 66x.
//  A (16x4 f32 per K-chunk)  = 16 gathered sample rows, reloaded per slot.
//  32 x v_wmma_f32_16x16x4_f32 accumulate the K=128 dot products;
//  diagonal C(i,i) = dot(sample_i, anchor_i).
__global__ __launch_bounds__(256) void idloss_main_kernel(
    const float* __restrict__ ids_fut,
    const float* __restrict__ ids_hist,
    const int*   __restrict__ anchor_idx,
    const int*   __restrict__ pos_idx,
    const int*   __restrict__ neg_idx,
    float*       __restrict__ tile_out)
{
    const int lane = threadIdx.x & 31;
    const int wave = threadIdx.x >> 5;
    const int tile = blockIdx.x * 8 + wave;          // 0..1023 (exact grid)
    const int l    = lane & 15;                      // anchor slot in tile
    const int hi   = lane >> 4;                      // 0: K={0,1}, 1: K={2,3} per chunk
    const int a    = tile * 16 + l;                  // global anchor id

    // ---- B fragments: anchor columns (ids_hist gather), plus anchor norm ----
    const int    hrow  = anchor_idx[a];
    const float* hbase = ids_hist + (size_t)hrow * D_DIM + 2 * hi;
    v2f   bfrag[32];
    float napart = 0.f;
#pragma unroll
    for (int c = 0; c < 32; ++c) {
        bfrag[c] = *(const v2f*)(hbase + 4 * c);     // global_load_b64 gather
        napart  += bfrag[c].x * bfrag[c].x + bfrag[c].y * bfrag[c].y;
    }
    // row l's 128 elements live on lanes l and l+16 -> combine halves
    const float na = sqrtf(napart + __shfl_xor(napart, 16, 32));

    float sp = 0.f, sn = 0.f;

    for (int s = 0; s < P_POS + N_NEG; ++s) {        // uniform loop: EXEC stays all-1
        const int srow = (s < P_POS) ? pos_idx[a * P_POS + s]
                                     : neg_idx[a * N_NEG + (s - P_POS)];
        const float* sbase = ids_fut + (size_t)srow * D_DIM + 2 * hi;

        v2f   afrag[32];
        float nspart = 0.f;
#pragma unroll
        for (int c = 0; c < 32; ++c) {
            afrag[c] = *(const v2f*)(sbase + 4 * c); // global_load_b64 gather
            nspart  += afrag[c].x * afrag[c].x + afrag[c].y * afrag[c].y;
        }
        const float ns = sqrtf(nspart + __shfl_xor(nspart, 16, 32));

        v8f cacc = {};
#pragma unroll
        for (int c = 0; c < 32; ++c) {
            // 8 args: (neg_a, A, neg_b, B, c_mod, C, reuse_a, reuse_b)
            cacc = __builtin_amdgcn_wmma_f32_16x16x4_f32(
                false, afrag[c], false, bfrag[c], (short)0, cacc, false, false);
        }

        // Diagonal extraction (f32 C layout: i<8 -> VGPR i lane i; i>=8 -> VGPR i-8 lane i+16)
        float d = 0.f;
#pragma unroll
        for (int r = 0; r < 8; ++r) {
            const float vlo = __shfl(cacc[r], r,      32);
            const float vhi = __shfl(cacc[r], r + 24, 32);
            if (lane == r)     d = vlo;
            if (lane == r + 8) d = vhi;
        }

        const float cosp1 = d / fmaxf(na * ns, EPS_V) + 1.0f;  // torch CosineSimilarity + 1
        if (s < P_POS) sp += cosp1; else sn += cosp1;
    }

    // lanes 0..15 hold valid per-anchor sums; -log(sp/(sp+sn)) = log(sp+sn) - log(sp)
    float loss = logf(sp + sn) - logf(sp);
    if (lane >= 16) loss = 0.f;
#pragma unroll
    for (int m = 8; m >= 1; m >>= 1)
        loss += __shfl_xor(loss, m, 32);             // sums lanes 0..15 into lane 0
    if (lane == 0) tile_out[tile] = loss;
}

// Deterministic fixed-order reduction of the 1024 tile partials -> d_out[0].
__global__ __launch_bounds__(256) void idloss_reduce_kernel(
    const float* __restrict__ tile_out, float* __restrict__ out)
{
    __shared__ float smem[256];
    const int t = threadIdx.x;
    float v = 0.f;
    for (int i = t; i < NTILES; i += 256) v += tile_out[i];   // fixed order
    smem[t] = v;
    __syncthreads();
    for (int k = 128; k > 0; k >>= 1) {
        if (t < k) smem[t] += smem[t + k];
        __syncthreads();
    }
    if (t == 0) out[0] = smem[0];
}

extern "C" void kernel_launch(void* const* d_in, const int* in_sizes, int n_in,
                              void* d_out, int out_size, void* d_ws, size_t ws_size,
                              hipStream_t stream) {
    (void)in_sizes; (void)n_in; (void)out_size; (void)ws_size;
    const float* ids_fut    = (const float*)d_in[0];
    const float* ids_hist   = (const float*)d_in[1];
    const int*   anchor_idx = (const int*)d_in[2];
    const int*   pos_idx    = (const int*)d_in[3];
    const int*   neg_idx    = (const int*)d_in[4];
    float* tile_out = (float*)d_ws;          // NTILES floats (4 KB) of scratch
    float* out      = (float*)d_out;

    // 1024 waves total: 128 blocks x 8 waves; one wave per 16-anchor tile.
    idloss_main_kernel<<<NTILES / 8, 256, 0, stream>>>(
        ids_fut, ids_hist, anchor_idx, pos_idx, neg_idx, tile_out);
    idloss_reduce_kernel<<<1, 256, 0, stream>>>(tile_out, out);
}